// KCGE_11527692223096
// MI455X (gfx1250) — compile-verified
//
#include <hip/hip_runtime.h>
#include <hip/hip_bf16.h>

#define DD 128            // feature dim
#define RR 8              // relations

typedef float v2f __attribute__((ext_vector_type(2)));
typedef float v8f __attribute__((ext_vector_type(8)));

// ---------------------------------------------------------------------------
// deg[col[e]] += 1
__global__ void k_deg(const int* __restrict__ col, float* __restrict__ deg, int E) {
    int e = blockIdx.x * blockDim.x + threadIdx.x;
    if (e < E) atomicAdd(&deg[col[e]], 1.0f);
}

// dis[n] = deg==0 ? 1 : deg^-0.5
__global__ void k_dis(const float* __restrict__ deg, float* __restrict__ dis, int N) {
    int n = blockIdx.x * blockDim.x + threadIdx.x;
    if (n < N) {
        float d = deg[n];
        dis[n] = (d == 0.0f) ? 1.0f : __frsqrt_rn(d);
    }
}

// coef[e] = dis[row]*dis[col]*w[e]
__global__ void k_coef(const int* __restrict__ row, const int* __restrict__ col,
                       const float* __restrict__ ew, const float* __restrict__ dis,
                       float* __restrict__ coef, int E) {
    int e = blockIdx.x * blockDim.x + threadIdx.x;
    if (e < E) coef[e] = dis[row[e]] * dis[col[e]] * ew[e];
}

// ---------------------------------------------------------------------------
// T_out[0:nrows,0:128] = X[0:nrows,0:128] @ W[rbase + blockIdx.y]
// (fp32 WMMA 16x16x4). Block: 256 thr = 8 waves; covers 128 rows.
// Each wave: 16 rows x 128 cols = 8 accumulator tiles.
//
// LDS holds W in K-PAIR-INTERLEAVED layout:
//     wlds[(k>>1)*256 + 2*n + (k&1)]  ==  W[k][n]
// so a B fragment (W[ks][n], W[ks+1][n]) with even ks is one contiguous
// 8-byte read -> single ds_load_b64 into an aligned VGPR pair (no movs).
__global__ __launch_bounds__(256) void k_gemm_xw(const float* __restrict__ X,
                                                 const float* __restrict__ W,
                                                 float* __restrict__ T,
                                                 int nrows, int rbase,
                                                 size_t tStride) {
    __shared__ float wlds[DD * DD];                       // 64 KB of 320 KB WGP LDS
    const int r = rbase + blockIdx.y;
    const float* Wr = W + (size_t)r * DD * DD;
    float* Tr = T + (size_t)blockIdx.y * tStride;

    // Stage W with interleave: work item j covers row-pair kp = j/32,
    // col group n0 = (j%32)*4.  2x global b128 in, 2x ds_store_b128 out.
    for (int j = threadIdx.x; j < (DD / 2) * (DD / 4); j += 256) {
        const int kp = j >> 5;
        const int n0 = (j & 31) << 2;
        float4 g0 = *(const float4*)(Wr + (size_t)(2 * kp) * DD + n0);
        float4 g1 = *(const float4*)(Wr + (size_t)(2 * kp + 1) * DD + n0);
        float* dst = wlds + kp * 256 + 2 * n0;
        ((float4*)dst)[0] = make_float4(g0.x, g1.x, g0.y, g1.y);
        ((float4*)dst)[1] = make_float4(g0.z, g1.z, g0.w, g1.w);
    }
    __syncthreads();

    const int wave = threadIdx.x >> 5;
    const int lane = threadIdx.x & 31;
    const int lo16 = lane & 15;
    const bool hi  = lane >= 16;

    const int m0   = blockIdx.x * 128 + wave * 16;
    const int mrow = min(m0 + lo16, nrows - 1);           // clamp: no divergence
    const float* xrow = X + (size_t)mrow * DD;

    v8f acc[8];
#pragma unroll
    for (int t = 0; t < 8; ++t) acc[t] = (v8f){0,0,0,0,0,0,0,0};

    // A 16x4 fp32: lane m=l%16, VGPR j holds K = k0 + j + (l>=16 ? 2 : 0)
    // B 4x16  fp32: lane n=l%16, VGPR j holds K = k0 + j + (l>=16 ? 2 : 0)
    for (int k0 = 0; k0 < DD; k0 += 4) {
        const int ks = k0 + (hi ? 2 : 0);                 // always even
        v2f a = *(const v2f*)(xrow + ks);                 // contiguous -> b64 load
        const float* brow = wlds + (ks >> 1) * 256 + 2 * lo16;
#pragma unroll
        for (int t = 0; t < 8; ++t) {
            v2f b = *(const v2f*)(brow + t * 32);         // one ds_load_b64
            acc[t] = __builtin_amdgcn_wmma_f32_16x16x4_f32(
                /*neg_a=*/false, a, /*neg_b=*/false, b,
                /*c_mod=*/(short)0, acc[t], /*reuse_a=*/false, /*reuse_b=*/false);
        }
    }

    // C/D layout: VGPR v, lane l -> M = m0 + v + (l>=16?8:0), N = l%16
#pragma unroll
    for (int t = 0; t < 8; ++t) {
        const int n = t * 16 + lo16;
#pragma unroll
        for (int v = 0; v < 8; ++v) {
            const int m = m0 + v + (hi ? 8 : 0);
            if (m < nrows) Tr[(size_t)m * DD + n] = acc[t][v];
        }
    }
}

// ---------------------------------------------------------------------------
// FUSED: agg[col[e],:] += coef[e] * T[etype[e]][row[e],:]   (single pass)
// One wave per edge, 4 floats per lane (float4 gather + 4 float atomics).
__global__ __launch_bounds__(256) void k_edge_scatter_all(const int* __restrict__ row,
                                                          const int* __restrict__ col,
                                                          const int* __restrict__ etype,
                                                          const float* __restrict__ coef,
                                                          const float* __restrict__ T,
                                                          float* __restrict__ agg,
                                                          int E, size_t tStride) {
    const int wave = threadIdx.x >> 5;
    const int lane = threadIdx.x & 31;
    const int e = blockIdx.x * 8 + wave;                  // wave-uniform
    if (e >= E) return;
    const float c = coef[e];
    const int  rs = row[e];
    const int  cs = col[e];
    const int  t  = etype[e];
    const float* src = T + (size_t)t * tStride + (size_t)rs * DD + lane * 4;
    float4 v = *(const float4*)src;
    float* dst = agg + (size_t)cs * DD + lane * 4;
    atomicAdd(dst + 0, c * v.x);
    atomicAdd(dst + 1, c * v.y);
    atomicAdd(dst + 2, c * v.z);
    atomicAdd(dst + 3, c * v.w);
}

// FALLBACK: per-relation pass (small workspace): only edges with etype==r.
__global__ __launch_bounds__(256) void k_edge_scatter_r(const int* __restrict__ row,
                                                        const int* __restrict__ col,
                                                        const int* __restrict__ etype,
                                                        const float* __restrict__ coef,
                                                        const float* __restrict__ T,
                                                        float* __restrict__ agg,
                                                        int r, int E) {
    const int wave = threadIdx.x >> 5;
    const int lane = threadIdx.x & 31;
    const int e = blockIdx.x * 8 + wave;                  // wave-uniform
    if (e >= E) return;
    if (etype[e] != r) return;                            // wave-uniform predicate
    const float c = coef[e];
    const int  rs = row[e];
    const int  cs = col[e];
    float4 v = *(const float4*)(T + (size_t)rs * DD + lane * 4);
    float* dst = agg + (size_t)cs * DD + lane * 4;
    atomicAdd(dst + 0, c * v.x);
    atomicAdd(dst + 1, c * v.y);
    atomicAdd(dst + 2, c * v.z);
    atomicAdd(dst + 3, c * v.w);
}

// z1 = leaky_relu(agg + b, 0.01)
__global__ void k_bias_lrelu(const float* __restrict__ agg, const float* __restrict__ b,
                             float* __restrict__ z1, int total) {
    int i = blockIdx.x * blockDim.x + threadIdx.x;
    if (i < total) {
        float v = agg[i] + b[i & (DD - 1)];
        z1[i] = (v > 0.0f) ? v : 0.01f * v;
    }
}

// z2 = agg2 + b2; out[0:ND)=(x+z1+z2)/4 ; out[ND:2ND)=(z1+z2)/3
__global__ void k_final(const float* __restrict__ x, const float* __restrict__ z1,
                        const float* __restrict__ agg2, const float* __restrict__ b2,
                        float* __restrict__ out, size_t ND) {
    size_t i = (size_t)blockIdx.x * blockDim.x + threadIdx.x;
    if (i < ND) {
        float z2 = agg2[i] + b2[i & (DD - 1)];
        float a  = z1[i];
        out[i]      = (x[i] + a + z2) * 0.25f;
        out[ND + i] = (a + z2) * (1.0f / 3.0f);
    }
}

// ---------------------------------------------------------------------------
extern "C" void kernel_launch(void* const* d_in, const int* in_sizes, int n_in,
                              void* d_out, int out_size, void* d_ws, size_t ws_size,
                              hipStream_t stream) {
    const float* x     = (const float*)d_in[0];
    const int*   ei    = (const int*)d_in[1];   // [2,E] flat: row then col
    const int*   etype = (const int*)d_in[2];
    const float* ew    = (const float*)d_in[3];
    const float* W1    = (const float*)d_in[4];
    const float* b1    = (const float*)d_in[5];
    const float* W2    = (const float*)d_in[6];
    const float* b2    = (const float*)d_in[7];
    float* out = (float*)d_out;

    const int N = in_sizes[0] / DD;            // 50000
    const int E = in_sizes[3];                 // 640000
    const size_t ND = (size_t)N * DD;

    const int* row = ei;
    const int* col = ei + E;

    // workspace layout (floats): deg[N] dis[N] coef[E] T[...] agg[ND] z1[ND]
    float* ws   = (float*)d_ws;
    float* deg  = ws;
    float* dis  = deg  + N;
    float* coef = dis  + N;
    float* T    = coef + E;

    const size_t fixed = (size_t)2 * N + E + 2 * ND;      // everything except T
    const bool fused = ws_size >= (fixed + (size_t)RR * ND) * sizeof(float);
    const size_t tFloats = fused ? (size_t)RR * ND : ND;

    float* agg  = T + tFloats;
    float* z1   = agg + ND;

    hipMemsetAsync(deg, 0, (size_t)N * sizeof(float), stream);
    hipMemsetAsync(agg, 0, ND * sizeof(float), stream);

    const int t256_E = (E + 255) / 256;
    const int t256_N = (N + 255) / 256;
    k_deg<<<t256_E, 256, 0, stream>>>(col, deg, E);
    k_dis<<<t256_N, 256, 0, stream>>>(deg, dis, N);
    k_coef<<<t256_E, 256, 0, stream>>>(row, col, ew, dis, coef, E);

    const int gemmBlocksX = (N + 127) / 128;
    const int edgeBlocks  = (E + 7) / 8;
    const int ndBlocks    = (int)((ND + 255) / 256);

    if (fused) {
        dim3 gg(gemmBlocksX, RR);
        // ---- conv 1 ----
        k_gemm_xw<<<gg, 256, 0, stream>>>(x, W1, T, N, 0, ND);
        k_edge_scatter_all<<<edgeBlocks, 256, 0, stream>>>(row, col, etype, coef, T, agg, E, ND);
        k_bias_lrelu<<<ndBlocks, 256, 0, stream>>>(agg, b1, z1, (int)ND);
        // ---- conv 2 ----
        hipMemsetAsync(agg, 0, ND * sizeof(float), stream);
        k_gemm_xw<<<gg, 256, 0, stream>>>(z1, W2, T, N, 0, ND);
        k_edge_scatter_all<<<edgeBlocks, 256, 0, stream>>>(row, col, etype, coef, T, agg, E, ND);
    } else {
        // ---- conv 1 ----
        for (int r = 0; r < RR; ++r) {
            k_gemm_xw<<<dim3(gemmBlocksX, 1), 256, 0, stream>>>(x, W1, T, N, r, 0);
            k_edge_scatter_r<<<edgeBlocks, 256, 0, stream>>>(row, col, etype, coef, T, agg, r, E);
        }
        k_bias_lrelu<<<ndBlocks, 256, 0, stream>>>(agg, b1, z1, (int)ND);
        // ---- conv 2 ----
        hipMemsetAsync(agg, 0, ND * sizeof(float), stream);
        for (int r = 0; r < RR; ++r) {
            k_gemm_xw<<<dim3(gemmBlocksX, 1), 256, 0, stream>>>(z1, W2, T, N, r, 0);
            k_edge_scatter_r<<<edgeBlocks, 256, 0, stream>>>(row, col, etype, coef, T, agg, r, E);
        }
    }

    k_final<<<ndBlocks, 256, 0, stream>>>(x, z1, agg, b2, out, ND);
}